// GIN_classifier_to_explain_54322746360001
// MI455X (gfx1250) — compile-verified
//
#include <hip/hip_runtime.h>
#include <hip/hip_bf16.h>

#define NNODES 65536
#define CFEAT  128
#define NEDGE  1048576
#define NGRAPH 64
#define NPG    1024
#define SLOPE  0.01f

typedef __attribute__((ext_vector_type(2))) float v2f;
typedef __attribute__((ext_vector_type(8))) float v8f;

__device__ __forceinline__ float leaky(float x) { return x >= 0.f ? x : SLOPE * x; }

// out = in (float4 copy); initializes the aggregation buffer with h so the
// edge kernel directly produces h + segment_sum(h[src], dst).
__global__ void copy_kernel(const float* __restrict__ in, float* __restrict__ out, int n4) {
    int i = blockIdx.x * blockDim.x + threadIdx.x;
    if (i < n4) ((float4*)out)[i] = ((const float4*)in)[i];
}

// 32 threads per edge, 4 floats per thread: coalesced 512B row gather + fp32 atomics.
// h (32MB) is L2-resident on MI455X (192MB L2), so this runs at L2 speed.
__global__ void edge_kernel(const float* __restrict__ h,
                            const int* __restrict__ srcs,
                            const int* __restrict__ dsts,
                            float* __restrict__ agg) {
    unsigned gid = blockIdx.x * blockDim.x + threadIdx.x;
    unsigned e = gid >> 5;
    int c = (gid & 31) * 4;
    int s = srcs[e];
    int d = dsts[e];
    float4 v = *(const float4*)(h + (size_t)s * CFEAT + c);
    float* p = agg + (size_t)d * CFEAT + c;
    atomicAdd(p + 0, v.x);
    atomicAdd(p + 1, v.y);
    atomicAdd(p + 2, v.z);
    atomicAdd(p + 3, v.w);
}

// out[N x 128] = act(A[N x 128] @ W[128 x 128] + bias), act: 0=none, 1=leaky.
// One wave per 16-row stripe; fp32 WMMA 16x16x4 (exact fp32 math).
// W is staged once per block into LDS in B-fragment order so each fragment is a
// single conflict-free ds_load_b64 (8x less L2 traffic than per-wave global loads).
__global__ __launch_bounds__(256) void gemm_kernel(const float* __restrict__ A,
                                                   const float* __restrict__ W,
                                                   const float* __restrict__ bias,
                                                   float* __restrict__ out,
                                                   int act) {
    // slot s = kk*256 + nt*32 + lane holds
    //   (W[4kk + 2*(lane>>4)    ][16nt + (lane&15)],
    //    W[4kk + 2*(lane>>4) + 1][16nt + (lane&15)])
    __shared__ v2f wlds[32 * 8 * 32];   // 64 KB

    int tid  = threadIdx.x;
    int lane = tid & 31;
    int wv   = tid >> 5;

    {   // Stage W -> LDS, fragment-ordered. tid encodes (nt = tid>>5, lane = tid&31).
        int krow_b = 2 * (lane >> 4);
        int coln   = 16 * wv + (lane & 15);
#pragma unroll
        for (int kk = 0; kk < 32; ++kk) {
            int krow = 4 * kk + krow_b;
            v2f f;
            f.x = W[krow * CFEAT + coln];
            f.y = W[(krow + 1) * CFEAT + coln];
            wlds[kk * 256 + tid] = f;
        }
    }

    int r0   = blockIdx.x * 128 + wv * 16;
    int rA   = r0 + (lane & 15);
    int koff = 2 * (lane >> 4);     // lanes 0-15 -> K={0,1}; lanes 16-31 -> K={2,3}
    int col  = lane & 15;

    // Load all 32 A-fragments (K in steps of 4) for this wave's 16-row stripe.
    v2f a[32];
    const float* arow = A + (size_t)rA * CFEAT + koff;
#pragma unroll
    for (int kk = 0; kk < 32; ++kk) a[kk] = *(const v2f*)(arow + kk * 4);

    __syncthreads();

#pragma unroll 1
    for (int nt = 0; nt < 8; ++nt) {
        v8f acc = {};
        const v2f* wf = &wlds[nt * 32 + lane];
#pragma unroll
        for (int kk = 0; kk < 32; ++kk) {
            acc = __builtin_amdgcn_wmma_f32_16x16x4_f32(
                false, a[kk], false, wf[kk * 256], (short)0, acc, false, false);
        }
        float bv = bias[nt * 16 + col];
        int rowBase = r0 + 8 * (lane >> 4);
#pragma unroll
        for (int v = 0; v < 8; ++v) {
            float val = acc[v] + bv;
            if (act) val = leaky(val);
            out[(size_t)(rowBase + v) * CFEAT + nt * 16 + col] = val;
        }
    }
}

// y[n] = dot(leaky(h[n,:]), fc1_w) + fc1_b   (one wave per node)
__global__ void fc1_kernel(const float* __restrict__ h,
                           const float* __restrict__ w,
                           const float* __restrict__ b,
                           float* __restrict__ y) {
    int tid  = blockIdx.x * blockDim.x + threadIdx.x;
    int node = tid >> 5;
    int lane = tid & 31;
    float4 hv = *(const float4*)(h + (size_t)node * CFEAT + lane * 4);
    float4 wv = *(const float4*)(w + lane * 4);
    float s = leaky(hv.x) * wv.x + leaky(hv.y) * wv.y +
              leaky(hv.z) * wv.z + leaky(hv.w) * wv.w;
    for (int off = 16; off > 0; off >>= 1) s += __shfl_xor(s, off, 32);
    if (lane == 0) y[node] = s + b[0];
}

// logits[g,k] = sum_n leaky(y[g,n]) * fc2_w[k,n] + fc2_b[k]; then log_softmax.
__global__ void fc2_kernel(const float* __restrict__ y,
                           const float* __restrict__ w,
                           const float* __restrict__ b,
                           float* __restrict__ out) {
    int g = blockIdx.x;
    int tid = threadIdx.x;
    float s0 = 0.f, s1 = 0.f;
    for (int n = tid; n < NPG; n += 256) {
        float v = leaky(y[(size_t)g * NPG + n]);
        s0 += v * w[n];
        s1 += v * w[NPG + n];
    }
    __shared__ float r0[256], r1[256];
    r0[tid] = s0; r1[tid] = s1;
    __syncthreads();
    for (int st = 128; st > 0; st >>= 1) {
        if (tid < st) { r0[tid] += r0[tid + st]; r1[tid] += r1[tid + st]; }
        __syncthreads();
    }
    if (tid == 0) {
        float l0 = r0[0] + b[0];
        float l1 = r1[0] + b[1];
        float m = fmaxf(l0, l1);
        float lse = m + logf(expf(l0 - m) + expf(l1 - m));
        out[g * 2 + 0] = l0 - lse;
        out[g * 2 + 1] = l1 - lse;
    }
}

extern "C" void kernel_launch(void* const* d_in, const int* in_sizes, int n_in,
                              void* d_out, int out_size, void* d_ws, size_t ws_size,
                              hipStream_t stream) {
    const float* x    = (const float*)d_in[0];
    const float* W1   = (const float*)d_in[1];
    const float* b1   = (const float*)d_in[2];
    const float* W2   = (const float*)d_in[3];
    const float* b2   = (const float*)d_in[4];
    const float* fc1w = (const float*)d_in[5];
    const float* fc1b = (const float*)d_in[6];
    const float* fc2w = (const float*)d_in[7];
    const float* fc2b = (const float*)d_in[8];
    const int*   ei   = (const int*)d_in[9];   // [2, E]: src row then dst row
    // d_in[10] = batch (unused; reference pools by reshape order)

    float* bufA = (float*)d_ws;                      // 32 MB
    float* bufB = bufA + (size_t)NNODES * CFEAT;     // 32 MB

    const int n4 = NNODES * CFEAT / 4;
    const float* h = x;
    for (int l = 0; l < 4; ++l) {
        float* aggb = (l & 1) ? bufA : bufB;
        float* tb   = (l & 1) ? bufB : bufA;
        copy_kernel<<<n4 / 256, 256, 0, stream>>>(h, aggb, n4);
        edge_kernel<<<(NEDGE * 32) / 256, 256, 0, stream>>>(h, ei, ei + NEDGE, aggb);
        gemm_kernel<<<NNODES / 128, 256, 0, stream>>>(
            aggb, W1 + (size_t)l * CFEAT * CFEAT, b1 + l * CFEAT, tb, 1);
        gemm_kernel<<<NNODES / 128, 256, 0, stream>>>(
            tb, W2 + (size_t)l * CFEAT * CFEAT, b2 + l * CFEAT, aggb, (l < 3) ? 1 : 0);
        h = aggb;
    }
    // After layer 3, h == bufA; bufB is dead -> reuse for the per-node FC1 output.
    float* ybuf = bufB;
    fc1_kernel<<<(NNODES * 32) / 256, 256, 0, stream>>>(h, fc1w, fc1b, ybuf);
    fc2_kernel<<<NGRAPH, 256, 0, stream>>>(ybuf, fc2w, fc2b, (float*)d_out);
}